// GraphModel_76982993813741
// MI455X (gfx1250) — compile-verified
//
#include <hip/hip_runtime.h>
#include <hip/hip_bf16.h>
#include <math.h>

typedef float v2f __attribute__((ext_vector_type(2)));
typedef float v8f __attribute__((ext_vector_type(8)));

#define N_NODES 50000
#define N_EDGES 800000
#define HID 64
#define HEADS 4
#define HC (HEADS * HID)   // 256

// ---------------------------------------------------------------------------
// float atomic max via sign-split int/uint monotonic mapping (always compiles,
// correct for mixed-sign values; amax initialized to -FLT_MAX).
// ---------------------------------------------------------------------------
__device__ __forceinline__ void atomicMaxFloat(float* addr, float val) {
    if (val >= 0.0f) {
        atomicMax((int*)addr, __float_as_int(val));
    } else {
        atomicMin((unsigned int*)addr, (unsigned int)__float_as_int(val));
    }
}

__device__ __forceinline__ void edge_nodes(const int* __restrict__ ei, int e,
                                           int& s, int& d) {
    if (e < N_EDGES) { s = ei[e]; d = ei[N_EDGES + e]; }
    else             { s = d = e - N_EDGES; }          // appended self-loops
}

// ---------------------------------------------------------------------------
// GEMM1: h[N,256] = x[N,64] @ W_gat[64,256]   (fp32 WMMA 16x16x4)
// Block = 256 threads = 8 waves; block owns a 16-row M tile; wave w owns
// two column tiles [32w, 32w+32). A tile staged in LDS; W_gat (64KB) is L2-hot.
// ---------------------------------------------------------------------------
__global__ void __launch_bounds__(256) gemm_h_kernel(const float* __restrict__ x,
                                                     const float* __restrict__ W,
                                                     float* __restrict__ h) {
    __shared__ float sA[16 * 64];                       // 4 KB x-tile, row major
    const int m0   = blockIdx.x * 16;
    const int tid  = threadIdx.x;
    const int wave = tid >> 5;
    const int lane = tid & 31;

    for (int i = tid; i < 16 * 64; i += 256) {
        int r = i >> 6, c = i & 63;
        sA[i] = x[(size_t)(m0 + r) * HID + c];
    }
    __syncthreads();

    const int lm = lane & 15;          // M (for A) / N (for B,C,D) within tile
    const int kb = (lane >> 4) * 2;    // K pair base per ISA A-layout
    const int n0 = wave * 32;

    v8f acc0 = {}; v8f acc1 = {};
    for (int k0 = 0; k0 < 64; k0 += 4) {
        v2f a;
        a.x = sA[lm * 64 + k0 + kb];
        a.y = sA[lm * 64 + k0 + kb + 1];
        const float* Wk = W + (size_t)(k0 + kb) * HC;
        v2f b0, b1;
        b0.x = Wk[n0 + lm];       b0.y = Wk[HC + n0 + lm];
        b1.x = Wk[n0 + 16 + lm];  b1.y = Wk[HC + n0 + 16 + lm];
        acc0 = __builtin_amdgcn_wmma_f32_16x16x4_f32(false, a, false, b0,
                                                     (short)0, acc0, false, false);
        acc1 = __builtin_amdgcn_wmma_f32_16x16x4_f32(false, a, false, b1,
                                                     (short)0, acc1, false, false);
    }

    const int mrow = (lane >> 4) * 8;  // C/D: VGPR v -> rows v and v+8
    for (int v = 0; v < 8; ++v) {
        size_t m = (size_t)(m0 + mrow + v);
        h[m * HC + n0 + lm]      = acc0[v];
        h[m * HC + n0 + 16 + lm] = acc1[v];
    }
}

// ---------------------------------------------------------------------------
// Attention scores: a_src[n,h] = dot(h[n,h,:], att_src[h,:]); same for dst.
// One thread per (node, head); 256B contiguous per thread, L2-resident h.
// ---------------------------------------------------------------------------
__global__ void __launch_bounds__(256) att_kernel(const float* __restrict__ h,
                                                  const float* __restrict__ att_src,
                                                  const float* __restrict__ att_dst,
                                                  float* __restrict__ a_src,
                                                  float* __restrict__ a_dst) {
    int i = blockIdx.x * blockDim.x + threadIdx.x;      // i = node*HEADS + head
    if (i >= N_NODES * HEADS) return;
    int head = i & (HEADS - 1);
    const float* hp = h + (size_t)i * HID;
    const float* as = att_src + head * HID;
    const float* ad = att_dst + head * HID;
    float s = 0.f, d = 0.f;
    #pragma unroll 8
    for (int c = 0; c < HID; ++c) {
        float v = hp[c];
        s += v * as[c];
        d += v * ad[c];
    }
    a_src[i] = s;
    a_dst[i] = d;
}

// ---------------------------------------------------------------------------
// Init: accum = 0, amax = -FLT_MAX, denom = 0.
// ---------------------------------------------------------------------------
__global__ void __launch_bounds__(256) init_kernel(float* __restrict__ accum,
                                                   float* __restrict__ amax,
                                                   float* __restrict__ denom) {
    int i = blockIdx.x * blockDim.x + threadIdx.x;
    if (i < N_NODES * HC) accum[i] = 0.0f;
    if (i < N_NODES * HEADS) {
        amax[i]  = -3.402823466e+38f;
        denom[i] = 0.0f;
    }
}

// ---------------------------------------------------------------------------
// Edge pass 1: segment max of leaky_relu(a_src[src] + a_dst[dst]) by dst.
// ---------------------------------------------------------------------------
__global__ void __launch_bounds__(256) edge_max_kernel(const int* __restrict__ ei,
                                                       const float* __restrict__ a_src,
                                                       const float* __restrict__ a_dst,
                                                       float* __restrict__ amax) {
    int idx = blockIdx.x * blockDim.x + threadIdx.x;
    const int total = (N_EDGES + N_NODES) * HEADS;
    if (idx >= total) return;
    int e = idx >> 2, head = idx & 3;
    int s, d; edge_nodes(ei, e, s, d);
    float al = a_src[s * HEADS + head] + a_dst[d * HEADS + head];
    al = (al > 0.0f) ? al : 0.2f * al;
    atomicMaxFloat(&amax[d * HEADS + head], al);
}

// ---------------------------------------------------------------------------
// Edge pass 2: denom[dst] += exp(alpha - amax[dst]).
// ---------------------------------------------------------------------------
__global__ void __launch_bounds__(256) edge_sum_kernel(const int* __restrict__ ei,
                                                       const float* __restrict__ a_src,
                                                       const float* __restrict__ a_dst,
                                                       const float* __restrict__ amax,
                                                       float* __restrict__ denom) {
    int idx = blockIdx.x * blockDim.x + threadIdx.x;
    const int total = (N_EDGES + N_NODES) * HEADS;
    if (idx >= total) return;
    int e = idx >> 2, head = idx & 3;
    int s, d; edge_nodes(ei, e, s, d);
    float al = a_src[s * HEADS + head] + a_dst[d * HEADS + head];
    al = (al > 0.0f) ? al : 0.2f * al;
    float ex = expf(al - amax[d * HEADS + head]);
    atomicAdd(&denom[d * HEADS + head], ex);
}

// ---------------------------------------------------------------------------
// Edge pass 3: accum[dst, c] += h[src, c] * coef(head).  One block per edge,
// thread = channel (fully coalesced 1KB gather + 1KB atomic scatter, L2-hot).
// ---------------------------------------------------------------------------
__global__ void __launch_bounds__(256) edge_aggr_kernel(const int* __restrict__ ei,
                                                        const float* __restrict__ h,
                                                        const float* __restrict__ a_src,
                                                        const float* __restrict__ a_dst,
                                                        const float* __restrict__ amax,
                                                        const float* __restrict__ denom,
                                                        float* __restrict__ accum) {
    int e = blockIdx.x;
    int c = threadIdx.x;                  // 0..255
    int head = c >> 6;
    int s, d; edge_nodes(ei, e, s, d);
    float al = a_src[s * HEADS + head] + a_dst[d * HEADS + head];
    al = (al > 0.0f) ? al : 0.2f * al;
    float coef = expf(al - amax[d * HEADS + head]) /
                 (denom[d * HEADS + head] + 1e-16f);
    atomicAdd(&accum[(size_t)d * HC + c], h[(size_t)s * HC + c] * coef);
}

// ---------------------------------------------------------------------------
// GEMM2: out[N,64] = elu(accum + bias_gat) @ W_lin[256,64] + b_lin
// Block = 128 threads = 4 waves; wave w owns N tile w (N=64 -> 4 tiles).
// Bias + ELU fused into the LDS staging of the A tile.
// ---------------------------------------------------------------------------
__global__ void __launch_bounds__(128) gemm_out_kernel(const float* __restrict__ accum,
                                                       const float* __restrict__ bias,
                                                       const float* __restrict__ Wl,
                                                       const float* __restrict__ bl,
                                                       float* __restrict__ out) {
    __shared__ float sA[16 * HC];                       // 16 KB activation tile
    const int m0   = blockIdx.x * 16;
    const int tid  = threadIdx.x;
    const int wave = tid >> 5;
    const int lane = tid & 31;

    for (int i = tid; i < 16 * HC; i += 128) {
        int r = i >> 8, c = i & (HC - 1);
        float v = accum[(size_t)(m0 + r) * HC + c] + bias[c];
        sA[i] = (v > 0.0f) ? v : expm1f(v);             // ELU(alpha=1)
    }
    __syncthreads();

    const int lm = lane & 15;
    const int kb = (lane >> 4) * 2;
    const int n0 = wave * 16;

    v8f acc = {};
    for (int k0 = 0; k0 < HC; k0 += 4) {
        v2f a;
        a.x = sA[lm * HC + k0 + kb];
        a.y = sA[lm * HC + k0 + kb + 1];
        v2f b;
        b.x = Wl[(size_t)(k0 + kb) * HID + n0 + lm];
        b.y = Wl[(size_t)(k0 + kb + 1) * HID + n0 + lm];
        acc = __builtin_amdgcn_wmma_f32_16x16x4_f32(false, a, false, b,
                                                    (short)0, acc, false, false);
    }

    const float bn   = bl[n0 + lm];
    const int   mrow = (lane >> 4) * 8;
    for (int v = 0; v < 8; ++v)
        out[(size_t)(m0 + mrow + v) * HID + n0 + lm] = acc[v] + bn;
}

// ---------------------------------------------------------------------------
extern "C" void kernel_launch(void* const* d_in, const int* in_sizes, int n_in,
                              void* d_out, int out_size, void* d_ws, size_t ws_size,
                              hipStream_t stream) {
    const float* x        = (const float*)d_in[0];
    const int*   ei       = (const int*)  d_in[1];
    const float* W_gat    = (const float*)d_in[2];
    const float* att_src  = (const float*)d_in[3];
    const float* att_dst  = (const float*)d_in[4];
    const float* bias_gat = (const float*)d_in[5];
    const float* W_lin    = (const float*)d_in[6];
    const float* b_lin    = (const float*)d_in[7];
    float*       out      = (float*)d_out;

    // Workspace layout (~105.6 MB)
    float* h     = (float*)d_ws;                         // N*256
    float* accum = h     + (size_t)N_NODES * HC;         // N*256
    float* a_src = accum + (size_t)N_NODES * HC;         // N*4
    float* a_dst = a_src + (size_t)N_NODES * HEADS;
    float* amax  = a_dst + (size_t)N_NODES * HEADS;
    float* denom = amax  + (size_t)N_NODES * HEADS;

    const int ET = N_EDGES + N_NODES;                    // edges + self-loops

    gemm_h_kernel<<<N_NODES / 16, 256, 0, stream>>>(x, W_gat, h);
    att_kernel<<<(N_NODES * HEADS + 255) / 256, 256, 0, stream>>>(
        h, att_src, att_dst, a_src, a_dst);
    init_kernel<<<(N_NODES * HC + 255) / 256, 256, 0, stream>>>(accum, amax, denom);
    edge_max_kernel<<<(ET * HEADS + 255) / 256, 256, 0, stream>>>(
        ei, a_src, a_dst, amax);
    edge_sum_kernel<<<(ET * HEADS + 255) / 256, 256, 0, stream>>>(
        ei, a_src, a_dst, amax, denom);
    edge_aggr_kernel<<<ET, 256, 0, stream>>>(
        ei, h, a_src, a_dst, amax, denom, accum);
    gemm_out_kernel<<<N_NODES / 16, 128, 0, stream>>>(
        accum, bias_gat, W_lin, b_lin, out);
}